// PrototypicalContrastiveLoss_85452669321568
// MI455X (gfx1250) — compile-verified
//
#include <hip/hip_runtime.h>
#include <math.h>

#define TEMPERATURE 0.3f
#define MOMENTUM 0.5f
#define DDIM 64
#define MAXC 200

typedef __attribute__((ext_vector_type(2))) float v2f;
typedef __attribute__((ext_vector_type(8))) float v8f;

__device__ __forceinline__ float waveReduceSum32(float v) {
#pragma unroll
  for (int off = 16; off > 0; off >>= 1) v += __shfl_xor(v, off, 32);
  return v;
}
__device__ __forceinline__ float groupReduceSum16(float v) {
#pragma unroll
  for (int off = 8; off > 0; off >>= 1) v += __shfl_xor(v, off, 16);
  return v;
}

// ---------------------------------------------------------------------------
// Pass 1: per-row L2 normalize + LDS-privatized segment sums / counts.
// One wave per row: 32 lanes x float2 = 64 floats (256B coalesced per row).
// ---------------------------------------------------------------------------
__global__ void seg_accum_kernel(const float* __restrict__ feats,
                                 const int* __restrict__ labels,
                                 float* __restrict__ gsums,    // [C*64]
                                 float* __restrict__ gcounts,  // [C]
                                 int nRows, int C) {
  __shared__ float lsum[MAXC * DDIM];
  __shared__ float lcnt[MAXC];
  const int tid = threadIdx.x;
  const int nthr = blockDim.x;
  for (int i = tid; i < C * DDIM; i += nthr) lsum[i] = 0.f;
  for (int i = tid; i < C; i += nthr) lcnt[i] = 0.f;
  __syncthreads();

  const int lane = tid & 31;
  const int wave = tid >> 5;
  const int wavesPerBlock = nthr >> 5;
  const int waveGlobal = blockIdx.x * wavesPerBlock + wave;
  const int waveStride = gridDim.x * wavesPerBlock;

  for (int row = waveGlobal; row < nRows; row += waveStride) {
    v2f v = *(const v2f*)(feats + (size_t)row * DDIM + 2 * lane);
    float ss = waveReduceSum32(v.x * v.x + v.y * v.y);
    float inv = 1.0f / fmaxf(sqrtf(ss), 1e-12f);
    int lab = labels[row];
    float* dst = lsum + lab * DDIM + 2 * lane;
    atomicAdd(dst, v.x * inv);
    atomicAdd(dst + 1, v.y * inv);
    if (lane == 0) atomicAdd(&lcnt[lab], 1.0f);
  }
  __syncthreads();
  for (int i = tid; i < C * DDIM; i += nthr) atomicAdd(&gsums[i], lsum[i]);
  for (int i = tid; i < C; i += nthr) atomicAdd(&gcounts[i], lcnt[i]);
}

// ---------------------------------------------------------------------------
// Pass 2: prototype EMA finalize. One wave per class. Writes into a buffer
// padded to Ctiles*16 rows; padding rows stay zero (workspace is memset).
// ---------------------------------------------------------------------------
__global__ void proto_finalize_kernel(const float* __restrict__ gsums,
                                      const float* __restrict__ gcounts,
                                      const float* __restrict__ oldProto,
                                      float* __restrict__ newProto, int C) {
  const int c = blockIdx.x;
  const int lane = threadIdx.x;  // 32 threads
  if (c >= C) return;
  v2f s = *(const v2f*)(gsums + (size_t)c * DDIM + 2 * lane);
  float cnt = gcounts[c];
  float denom = fmaxf(cnt, 1.0f);
  float mx = s.x / denom, my = s.y / denom;
  float ss = waveReduceSum32(mx * mx + my * my);
  float inv = 1.0f / fmaxf(sqrtf(ss), 1e-12f);
  v2f old = *(const v2f*)(oldProto + (size_t)c * DDIM + 2 * lane);
  float bx = MOMENTUM * old.x + (1.f - MOMENTUM) * mx * inv;
  float by = MOMENTUM * old.y + (1.f - MOMENTUM) * my * inv;
  float ss2 = waveReduceSum32(bx * bx + by * by);
  float inv2 = 1.0f / fmaxf(sqrtf(ss2), 1e-12f);
  v2f out;
  if (cnt > 0.f) { out.x = bx * inv2; out.y = by * inv2; }
  else           { out = old; }
  *(v2f*)(newProto + (size_t)c * DDIM + 2 * lane) = out;
}

// ---------------------------------------------------------------------------
// Pass 3: loss. One wave owns 16 rows; logits via V_WMMA_F32_16X16X4_F32.
// proto is PADDED to Ctiles*16 zero rows -> B loads are unconditional. A
// sched_barrier pins all 16 global_load_b64 ahead of the WMMA chain (one
// clause, descending loadcnt drain) instead of load/wait/WMMA serialization.
// Unit-norm operands bound |logit| <= 1/T, so log-softmax needs no max
// subtraction; epilogue is branchless (select -1e30 before unconditional
// exp; label test cannot fire on padded columns since labels < C <= n).
// ---------------------------------------------------------------------------
#define LWAVES 4
#define TSTRIDE 68  // LDS row stride (floats) to spread banks

__global__ void __launch_bounds__(LWAVES * 32, 1)
proto_loss_kernel(const float* __restrict__ feats,
                  const int* __restrict__ labels,
                  const float* __restrict__ proto,  // padded [Ctiles*16][64]
                  float* __restrict__ accum,        // [0]=sum_nll [1]=count
                  int nRows, int C, int Ctiles) {
  __shared__ float tile[LWAVES][16 * TSTRIDE];
  const int tid = threadIdx.x;
  const int lane = tid & 31;
  const int wave = tid >> 5;
  const int rowBase = (blockIdx.x * LWAVES + wave) * 16;
  float* myTile = tile[wave];

  // stage 16 normalized rows into LDS (coalesced 256B row loads)
#pragma unroll
  for (int r = 0; r < 16; ++r) {
    int row = rowBase + r;
    v2f v = {0.f, 0.f};
    if (row < nRows) v = *(const v2f*)(feats + (size_t)row * DDIM + 2 * lane);
    float ss = waveReduceSum32(v.x * v.x + v.y * v.y);
    float inv = 1.0f / fmaxf(sqrtf(ss), 1e-12f);
    myTile[r * TSTRIDE + 2 * lane]     = v.x * inv;
    myTile[r * TSTRIDE + 2 * lane + 1] = v.y * inv;
  }
  __syncthreads();

  const int hi = lane >> 4;  // lane half selects K pair (A) / row half (C,D)
  const int lo = lane & 15;

  // A fragments for all 16 K-steps, reused across every class tile
  v2f afrag[16];
#pragma unroll
  for (int ks = 0; ks < 16; ++ks) {
    int kk = 4 * ks + 2 * hi;
    afrag[ks] = *(const v2f*)(myTile + lo * TSTRIDE + kk);
  }

  int lab[8], rowv[8];
#pragma unroll
  for (int r = 0; r < 8; ++r) {
    int row = rowBase + r + 8 * hi;
    rowv[r] = (row < nRows) ? 1 : 0;
    lab[r] = rowv[r] ? labels[row] : -1;
  }

  const float invT = 1.0f / TEMPERATURE;
  float s_part[8], l_part[8];  // per-lane partials (this lane's columns only)
#pragma unroll
  for (int r = 0; r < 8; ++r) { s_part[r] = 0.f; l_part[r] = 0.f; }

  for (int ct = 0; ct < Ctiles; ++ct) {
    const int n = ct * 16 + lo;
    const bool nvalid = (n < C);
    const float* bbase = proto + (size_t)n * DDIM + 2 * hi;

    // unconditional B fragments (padding rows are zero)
    v2f bfrag[16];
#pragma unroll
    for (int ks = 0; ks < 16; ++ks) bfrag[ks] = *(const v2f*)(bbase + 4 * ks);

    // keep all 16 loads ahead of the WMMA chain: one clause, 16 live regs
    __builtin_amdgcn_sched_barrier(0);

    v8f acc = {};
#pragma unroll
    for (int ks = 0; ks < 16; ++ks)
      acc = __builtin_amdgcn_wmma_f32_16x16x4_f32(false, afrag[ks], false,
                                                  bfrag[ks], (short)0, acc,
                                                  false, false);
#pragma unroll
    for (int r = 0; r < 8; ++r) {
      float v = acc[r] * invT;            // |v| <= 1/T: exp never overflows
      float vm = nvalid ? v : -1e30f;     // cndmask, then UNCONDITIONAL exp
      s_part[r] += __expf(vm);            // exp(-1e30) == 0 for padded cols
      l_part[r] += (n == lab[r]) ? v : 0.f;  // lab < C <= n on padded cols
    }
  }

  // one 16-lane reduction per row, once per wave
  float localSum = 0.f, localCnt = 0.f;
#pragma unroll
  for (int r = 0; r < 8; ++r) {
    float s = groupReduceSum16(s_part[r]);
    float l = groupReduceSum16(l_part[r]);
    if (lo == 0 && rowv[r] && lab[r] >= 0) {
      localSum += __logf(s) - l;
      localCnt += 1.f;
    }
  }
  if (lo == 0) {  // lanes 0 and 16: each owns 8 rows
    atomicAdd(&accum[0], localSum);
    atomicAdd(&accum[1], localCnt);
  }
}

__global__ void finalize_kernel(const float* __restrict__ acc,
                                float* __restrict__ out) {
  if (threadIdx.x == 0) {
    out[0] = acc[0] / fmaxf(acc[1], 1.0f);
    out[1] = acc[2] / fmaxf(acc[3], 1.0f);
  }
}

extern "C" void kernel_launch(void* const* d_in, const int* in_sizes, int n_in,
                              void* d_out, int out_size, void* d_ws, size_t ws_size,
                              hipStream_t stream) {
  (void)n_in; (void)out_size; (void)ws_size;
  const float* f_fti = (const float*)d_in[0];
  const float* e_rcl = (const float*)d_in[1];
  const int* lab_fti = (const int*)d_in[2];
  const int* lab_rcl = (const int*)d_in[3];
  const float* p_fti = (const float*)d_in[4];
  const float* p_rcl = (const float*)d_in[5];
  float* out = (float*)d_out;

  const int B  = in_sizes[0] / DDIM;  // 65536
  const int N  = in_sizes[1] / DDIM;  // 524288
  const int C1 = in_sizes[4] / DDIM;  // 50
  const int C2 = in_sizes[5] / DDIM;  // 200
  const int Ct1 = (C1 + 15) / 16, Ct2 = (C2 + 15) / 16;
  const int C1p = Ct1 * 16, C2p = Ct2 * 16;  // padded class counts

  float* ws    = (float*)d_ws;
  float* sums1 = ws;                   // C1*64
  float* cnts1 = sums1 + C1 * DDIM;    // C1
  float* sums2 = cnts1 + C1;           // C2*64
  float* cnts2 = sums2 + C2 * DDIM;    // C2
  float* np1   = cnts2 + C2;           // C1p*64 (padding rows stay zero)
  float* np2   = np1 + C1p * DDIM;     // C2p*64 (padding rows stay zero)
  float* acc   = np2 + C2p * DDIM;     // 4 floats
  size_t zeroBytes = (size_t)((char*)(acc + 4) - (char*)ws);
  hipMemsetAsync(d_ws, 0, zeroBytes, stream);

  seg_accum_kernel<<<256, 256, 0, stream>>>(f_fti, lab_fti, sums1, cnts1, B, C1);
  seg_accum_kernel<<<1024, 256, 0, stream>>>(e_rcl, lab_rcl, sums2, cnts2, N, C2);
  proto_finalize_kernel<<<C1, 32, 0, stream>>>(sums1, cnts1, p_fti, np1, C1);
  proto_finalize_kernel<<<C2, 32, 0, stream>>>(sums2, cnts2, p_rcl, np2, C2);

  proto_loss_kernel<<<(B + LWAVES * 16 - 1) / (LWAVES * 16), LWAVES * 32, 0, stream>>>(
      f_fti, lab_fti, np1, acc, B, C1, Ct1);
  proto_loss_kernel<<<(N + LWAVES * 16 - 1) / (LWAVES * 16), LWAVES * 32, 0, stream>>>(
      e_rcl, lab_rcl, np2, acc + 2, N, C2, Ct2);
  finalize_kernel<<<1, 32, 0, stream>>>(acc, out);
}